// PosEncoding_20624432955537
// MI455X (gfx1250) — compile-verified
//
#include <hip/hip_runtime.h>

// NeRF positional encoding, MI455X (gfx1250, wave32).
// Roofline: read 48MB + write 1.06GB => ~47us floor at 23.3 TB/s. Pure
// streaming op; WMMA is not applicable (no matrix contraction), so the
// CDNA5 paths used are async-to-LDS (ASYNCcnt) staging + NT b128 stores.
// Compute is minimal: per point exactly 30 v_ldexp + 30 v_sin + 30 v_cos
// (hardware sin/cos take "cycle" units: sin(2^i*pi*c) = v_sin_f32(2^(i-1)*c),
// args <= 256 so no range reduction), no integer division, no divergence.

#define BLOCK       256
#define TILE_PTS    256
#define OUT_PER_PT  63                       // 3 + 2*3*10
#define TILE_OUT    (TILE_PTS * OUT_PER_PT)  // 16128 floats = 64512 B LDS
#define TILE_V4     (TILE_OUT / 4)           // 4032 float4 per block
#define COORD_V4    ((TILE_PTS * 3) / 4)     // 192 float4 of coords

typedef float v4f __attribute__((ext_vector_type(4)));
typedef int   v4i __attribute__((ext_vector_type(4)));

typedef __attribute__((address_space(1))) v4i as1_v4i;
typedef __attribute__((address_space(3))) v4i as3_v4i;

#if __has_builtin(__builtin_amdgcn_global_load_async_to_lds_b128)
#define USE_ASYNC_LDS 1
#if __has_builtin(__builtin_amdgcn_s_wait_asynccnt)
#define WAIT_ASYNC() __builtin_amdgcn_s_wait_asynccnt(0)
#else
#define WAIT_ASYNC() asm volatile("s_wait_asynccnt 0" ::: "memory")
#endif
#else
#define USE_ASYNC_LDS 0
#endif

// Emit all 63 encoded values of one point. Fully unrolled: 30 ldexp, 30 sin,
// 30 cos, nothing else. Layout: [x,y,z, (i: (j: sin,cos))].
template <typename Emit>
__device__ __forceinline__ void encode_point(float x, float y, float z,
                                             Emit&& emit) {
    emit(0, x);
    emit(1, y);
    emit(2, z);
#pragma unroll
    for (int i = 0; i < 10; ++i) {
        float ax = __builtin_ldexpf(x, i - 1);   // v_ldexp_f32: 2^(i-1)*x
        float ay = __builtin_ldexpf(y, i - 1);
        float az = __builtin_ldexpf(z, i - 1);
        int b = 3 + i * 6;
        emit(b + 0, __builtin_amdgcn_sinf(ax));  // v_sin_f32 (cycle units)
        emit(b + 1, __builtin_amdgcn_cosf(ax));  // v_cos_f32
        emit(b + 2, __builtin_amdgcn_sinf(ay));
        emit(b + 3, __builtin_amdgcn_cosf(ay));
        emit(b + 4, __builtin_amdgcn_sinf(az));
        emit(b + 5, __builtin_amdgcn_cosf(az));
    }
}

__global__ __launch_bounds__(BLOCK)
void PosEncoding_20624432955537_kernel(const float* __restrict__ coords,
                                       float* __restrict__ out,
                                       long long npts) {
    __shared__ float sbuf[TILE_OUT];   // 64512 B; coords staged in its front
    const int t = threadIdx.x;
    const long long p0 = (long long)blockIdx.x * TILE_PTS;

    if (p0 + TILE_PTS <= npts) {
        // ---- stage 3KB coord tile into LDS (async DMA path) ----
#if USE_ASYNC_LDS
        if (t < COORD_V4) {            // waves 0..5, wave-uniform branch
            as1_v4i* g = (as1_v4i*)(coords + p0 * 3);
            as3_v4i* l = (as3_v4i*)sbuf;
            __builtin_amdgcn_global_load_async_to_lds_b128(g + t, l + t,
                                                           /*offset=*/0,
                                                           /*cpol=*/0);
            WAIT_ASYNC();
        }
#else
        if (t < COORD_V4) {
            ((v4f*)sbuf)[t] =
                __builtin_nontemporal_load((const v4f*)(coords + p0 * 3) + t);
        }
#endif
        __syncthreads();

        // ---- pull my point's coords (stride-3 dwords: conflict-free) ----
        float x = sbuf[t * 3 + 0];
        float y = sbuf[t * 3 + 1];
        float z = sbuf[t * 3 + 2];
        __syncthreads();               // reads done before overwrite

        // ---- compute 63 outputs into LDS at t*63 ----
        // stride 63 == -1 mod 64 banks -> each dword store is conflict-free.
        float* dst = sbuf + t * OUT_PER_PT;
        encode_point(x, y, z, [&](int k, float v) { dst[k] = v; });
        __syncthreads();

        // ---- coalesced NT drain: 4032 float4, 512B/contiguous per wave ----
        v4f* out4 = (v4f*)(out + (size_t)blockIdx.x * TILE_OUT);
        const v4f* s4 = (const v4f*)sbuf;
        for (int v = t; v < TILE_V4; v += BLOCK)
            __builtin_nontemporal_store(s4[v], out4 + v);
    } else {
        // ---- tail: one point per thread, direct scalar stores ----
        long long p = p0 + t;
        if (p < npts) {
            float x = coords[p * 3 + 0];
            float y = coords[p * 3 + 1];
            float z = coords[p * 3 + 2];
            float* dst = out + (size_t)p * OUT_PER_PT;
            encode_point(x, y, z, [&](int k, float v) { dst[k] = v; });
        }
    }
}

extern "C" void kernel_launch(void* const* d_in, const int* in_sizes, int n_in,
                              void* d_out, int out_size, void* d_ws, size_t ws_size,
                              hipStream_t stream) {
    (void)n_in; (void)out_size; (void)d_ws; (void)ws_size;
    const float* coords = (const float*)d_in[0];
    float* out = (float*)d_out;
    long long npts = (long long)in_sizes[0] / 3;
    int grid = (int)((npts + TILE_PTS - 1) / TILE_PTS);
    if (grid <= 0) return;
    PosEncoding_20624432955537_kernel<<<dim3(grid), dim3(BLOCK), 0, stream>>>(
        coords, out, npts);
}